// PlaneElement_80857054314544
// MI455X (gfx1250) — compile-verified
//
#include <hip/hip_runtime.h>
#include <stdint.h>

#define EPSF  1e-8f
#define BLOCK 256
#define CHUNK 2048

// x^(2/3) via hardware transcendentals: v_log_f32 (log2) + v_exp_f32 (exp2).
// ratio is always > 0 here (A_safe >= EPS, wp > 0), so log2 is well-defined.
__device__ __forceinline__ float pow23(float ratio) {
    return __builtin_amdgcn_exp2f(0.66666669f * __builtin_amdgcn_logf(ratio));
}

__device__ __forceinline__ float manning_q(float A, float WID, float invWID,
                                           float c12, float slman) {
    float h      = A * invWID;
    float wp     = fmaf(h, c12, WID);
    float A_safe = fmaxf(A, EPSF);
    float ratio  = A_safe / wp;
    return A * pow23(ratio) * slman;
}

__global__ __launch_bounds__(BLOCK) void plane_main(
    const float* __restrict__ depth,
    const float* __restrict__ p_rain,  const float* __restrict__ p_dt,
    const float* __restrict__ p_theta, const float* __restrict__ p_F,
    const float* __restrict__ p_WID,   const float* __restrict__ p_SS1,
    const float* __restrict__ p_SS2,   const float* __restrict__ p_MAN,
    const float* __restrict__ p_SL,    const float* __restrict__ p_dx,
    const float* __restrict__ p_Ks,    const float* __restrict__ p_psi,
    const float* __restrict__ p_thetas,
    float* __restrict__ ws, int n, int nb)
{
    // s_A layout: [2],[3] = left halo (g = gs-2, gs-1); [4 .. 4+CHUNK) = chunk
    //             [4+CHUNK] = right halo (g = gs+CHUNK). Offset 4 floats keeps
    //             the async b128 LDS destination 16-byte aligned.
    __shared__ __align__(16) float s_A[CHUNK + 5];
    __shared__ float s_flux[CHUNK + 1];   // flux for g in [gs-1, gs+CHUNK-1]

    const int tid = threadIdx.x;
    const int gs  = blockIdx.x * CHUNK;

    // uniform scalars (single-element input arrays)
    const float rain = *p_rain, dt = *p_dt, theta_c = *p_theta, Fc = *p_F;
    const float WID = *p_WID, SS1 = *p_SS1, SS2 = *p_SS2, MAN = *p_MAN;
    const float SL = *p_SL, dx = *p_dx, Ks = *p_Ks, psi = *p_psi, theta_s = *p_thetas;

    const float dtheta  = fmaxf(theta_s - theta_c, 0.0f);
    const float invF    = 1.0f / fmaxf(Fc, 1e-6f);
    const float rain_dt = rain * dt;
    const float c12     = sqrtf(1.0f + SS1 * SS1) + sqrtf(1.0f + SS2 * SS2);
    const float slman   = sqrtf(SL) / MAN;
    const float invWID  = 1.0f / WID;
    const float r       = dt / dx;
    const float psidt   = psi * dtheta;

    // ---- phase 1: async DMA of the chunk's depth into LDS (ASYNCcnt path) ----
    {
        uint32_t lds_base = (uint32_t)(uintptr_t)(&s_A[0]);
#pragma unroll
        for (int it = 0; it < CHUNK / (BLOCK * 4); ++it) {
            int j0 = (it * BLOCK + tid) * 4;                      // 4 floats / lane
            unsigned long long ga = (unsigned long long)(uintptr_t)(depth + gs + j0);
            uint32_t la = lds_base + (uint32_t)((4 + j0) * 4);    // 16B aligned
            asm volatile("global_load_async_to_lds_b128 %0, %1, off"
                         :: "v"(la), "v"(ga) : "memory");
        }
        if (tid < 2)  { int g = gs - 2 + tid; s_A[2 + tid]   = (g >= 0) ? depth[g] : 0.0f; }
        if (tid == 2) { int g = gs + CHUNK;   s_A[4 + CHUNK] = (g <  n) ? depth[g] : 0.0f; }
        asm volatile("s_wait_asynccnt 0" ::: "memory");
    }
    __syncthreads();

    // ---- phase 1b: Green-Ampt; depth -> A in place; accumulate infil sum ----
    float sum_infil = 0.0f;
    for (int m = tid; m < CHUNK + 3; m += BLOCK) {
        int   idx   = 2 + m;
        float d     = s_A[idx];
        float fp    = Ks * fmaf(psidt + d, invF, 1.0f);   // Ks*(1+(psi*dth+d)/F)
        float avail = rain_dt + d;
        float infil = fminf(fp * dt, avail);
        float surf  = fmaxf(d + rain_dt - infil, 0.0f);
        s_A[idx]    = surf * WID;
        if (idx >= 4 && idx < 4 + CHUNK) sum_infil += infil;   // owned nodes only
    }
    __syncthreads();

    // ---- phase 2: each face flux computed exactly once ----
    for (int j = tid; j <= CHUNK; j += BLOCK) {
        int   g  = gs - 1 + j;
        int   a  = j + 3;                 // LDS index for global node g
        float Ac = s_A[a];
        float slope = 0.0f;
        if (g > 0 && g < n - 1) {         // slope[0] = slope[N-1] = 0
            float dl = Ac - s_A[a - 1];
            float dr = s_A[a + 1] - Ac;
            slope = (dl * dr > 0.0f) ? ((fabsf(dl) < fabsf(dr)) ? dl : dr) : 0.0f;
        }
        float flux = 0.0f;
        if (g >= 0)
            flux = manning_q(fmaf(0.5f, slope, Ac), WID, invWID, c12, slman);
        s_flux[j] = flux;
    }
    __syncthreads();

    // ---- phase 3: upwind update, discharge, CFL max, outflow ----
    float max_cfl = 0.0f;
    for (int j = tid; j < CHUNK; j += BLOCK) {
        int   g  = gs + j;
        float Ac = s_A[j + 4];
        float fo = s_flux[j + 1];
        float fi = (g == 0) ? 0.0f : s_flux[j];
        float An = fmaxf(Ac - r * (fo - fi), 0.0f);
        float Q  = manning_q(An, WID, invWID, c12, slman);
        float vel = Q / fmaxf(An, EPSF);
        max_cfl = fmaxf(max_cfl, vel * r);
        if (g == n - 1) ws[2 * nb] = Q;               // outflow_q
    }
    __syncthreads();   // all LDS reads done before reusing s_A as scratch

    // ---- deterministic block reduction (serial-per-thread + fixed tree) ----
    s_A[tid]         = sum_infil;
    s_A[BLOCK + tid] = max_cfl;
    __syncthreads();
    for (int st = BLOCK / 2; st > 0; st >>= 1) {
        if (tid < st) {
            s_A[tid]         += s_A[tid + st];
            s_A[BLOCK + tid]  = fmaxf(s_A[BLOCK + tid], s_A[BLOCK + tid + st]);
        }
        __syncthreads();
    }
    if (tid == 0) {
        ws[blockIdx.x]      = s_A[0];       // partial infil-depth sum
        ws[nb + blockIdx.x] = s_A[BLOCK];   // partial cfl max
    }
}

__global__ __launch_bounds__(BLOCK) void plane_finalize(
    const float* __restrict__ ws, float* __restrict__ out,
    const float* __restrict__ p_dt, int n, int nb)
{
    __shared__ float ssum[BLOCK];
    __shared__ float smax[BLOCK];
    const int tid = threadIdx.x;
    float s = 0.0f, m = 0.0f;                 // cfl >= 0 so 0 is a safe identity
    for (int i = tid; i < nb; i += BLOCK) {
        s += ws[i];
        m  = fmaxf(m, ws[nb + i]);
    }
    ssum[tid] = s; smax[tid] = m;
    __syncthreads();
    for (int st = BLOCK / 2; st > 0; st >>= 1) {
        if (tid < st) {
            ssum[tid] += ssum[tid + st];
            smax[tid]  = fmaxf(smax[tid], smax[tid + st]);
        }
        __syncthreads();
    }
    if (tid == 0) {
        float dt  = *p_dt;
        float sum = ssum[0];
        float invN = 1.0f / (float)n;
        out[0] = ws[2 * nb];                  // outflow_q = Q[N-1]
        out[1] = sum * invN / dt;             // mean infil rate = mean depth / dt
        out[2] = sum * invN;                  // mean infil depth
        out[3] = smax[0];                     // max CFL
    }
}

extern "C" void kernel_launch(void* const* d_in, const int* in_sizes, int n_in,
                              void* d_out, int out_size, void* d_ws, size_t ws_size,
                              hipStream_t stream) {
    const float* depth    = (const float*)d_in[0];
    const float* p_rain   = (const float*)d_in[1];
    const float* p_dt     = (const float*)d_in[2];
    // d_in[3] = cum_rain: bookkeeping input, unused by the outputs
    const float* p_theta  = (const float*)d_in[4];
    const float* p_F      = (const float*)d_in[5];
    const float* p_WID    = (const float*)d_in[6];
    const float* p_SS1    = (const float*)d_in[7];
    const float* p_SS2    = (const float*)d_in[8];
    const float* p_MAN    = (const float*)d_in[9];
    const float* p_SL     = (const float*)d_in[10];
    const float* p_dx     = (const float*)d_in[11];
    const float* p_Ks     = (const float*)d_in[12];
    const float* p_psi    = (const float*)d_in[13];
    const float* p_thetas = (const float*)d_in[14];

    const int n  = in_sizes[0];          // 4,194,304 (divisible by CHUNK)
    const int nb = n / CHUNK;            // 2048 blocks
    float* ws = (float*)d_ws;            // [0,nb): sums, [nb,2nb): maxes, [2nb]: outflow

    plane_main<<<nb, BLOCK, 0, stream>>>(depth, p_rain, p_dt, p_theta, p_F,
                                         p_WID, p_SS1, p_SS2, p_MAN, p_SL,
                                         p_dx, p_Ks, p_psi, p_thetas,
                                         ws, n, nb);
    plane_finalize<<<1, BLOCK, 0, stream>>>(ws, (float*)d_out, p_dt, n, nb);
}